// GEMV_AWQ_11965778887017
// MI455X (gfx1250) — compile-verified
//
#include <hip/hip_runtime.h>
#include <hip/hip_bf16.h>

#define KDIM   4096
#define NDIM   14336
#define GRP    64
#define SPLIT  16            // K split for parallelism (57k threads in flight)
#define TPB    256
#define COLS   4             // columns per thread (int4 load per k)
#define COLS_PER_BLOCK (TPB * COLS)   // 1024
#define NB_N   (NDIM / COLS_PER_BLOCK) // 14

typedef unsigned int u32x4 __attribute__((ext_vector_type(4)));
typedef int          i32x8 __attribute__((ext_vector_type(8)));
typedef int          i32x4 __attribute__((ext_vector_type(4)));
typedef float        v2f   __attribute__((ext_vector_type(2)));
typedef float        v8f   __attribute__((ext_vector_type(8)));

__device__ __forceinline__ float bf_lo(unsigned u) {
  union { unsigned x; float f; } v; v.x = u << 16; return v.f;
}
__device__ __forceinline__ float bf_hi(unsigned u) {
  union { unsigned x; float f; } v; v.x = u & 0xFFFF0000u; return v.f;
}
__device__ __forceinline__ float bf1(unsigned short h) {
  union { unsigned x; float f; } v; v.x = (unsigned)h << 16; return v.f;
}

// D = x[1,K] * dequant(Q[K,N]); each block: 1024 columns x one K-chunk.
// SINGLE=true: whole K in one pass, writes bf16 out (+bias).
// SINGLE=false: writes f32 partials to ws[chunk][N].
template <bool SINGLE>
__global__ __launch_bounds__(TPB)
void awq_gemv_kernel(const __hip_bfloat16* __restrict__ x,
                     const int*            __restrict__ qw,
                     const __hip_bfloat16* __restrict__ sc,
                     const __hip_bfloat16* __restrict__ zr,
                     const __hip_bfloat16* __restrict__ bias,
                     __hip_bfloat16*       __restrict__ out,
                     float*                __restrict__ ws,
                     int chunk_len)
{
  __shared__ __attribute__((aligned(16))) unsigned short sx[KDIM]; // bf16 bits of x chunk
  __shared__ float sxsum[KDIM / GRP];                              // per-group sum of x

  const int chunk   = blockIdx.y;
  const int k0      = chunk * chunk_len;
  const int ngroups = chunk_len >> 6;   // chunk_len / 64

  // ---- Stage x[k0 .. k0+chunk_len) into LDS via the Tensor Data Mover ----
#if defined(__gfx1250__) && __has_builtin(__builtin_amdgcn_tensor_load_to_lds)
  if (threadIdx.x < 32) {  // one wave issues the DMA (TDM ignores EXEC)
    unsigned long long ga = (unsigned long long)(const void*)(x + k0);
    unsigned lds_off = (unsigned)(unsigned long long)(void*)&sx[0];
    u32x4 g0 = {0u, 0u, 0u, 0u};
    i32x8 g1 = {0, 0, 0, 0, 0, 0, 0, 0};
    i32x4 g2 = {0, 0, 0, 0};
    i32x4 g3 = {0, 0, 0, 0};
    i32x8 g4 = {0, 0, 0, 0, 0, 0, 0, 0};
    g0[0] = 1u;                                   // count = 1 valid descriptor
    g0[1] = lds_off;                              // lds_addr (bytes)
    g0[2] = (unsigned)ga;                         // global_addr[31:0]
    g0[3] = (unsigned)((ga >> 32) & 0x01FFFFFFull) | (2u << 30); // addr[56:32] | type=2
    g1[0] = (int)(1u << 16);                      // data_size = 2 bytes
    g1[1] = (int)(((unsigned)chunk_len & 0xFFFFu) << 16);  // tensor_dim0[15:0]
    g1[2] = (int)(((unsigned)chunk_len >> 16) | (1u << 16)); // dim0[31:16] | tensor_dim1=1
    g1[3] = (int)(((unsigned)chunk_len & 0xFFFFu) << 16);  // tile_dim0
    g1[4] = 1;                                    // tile_dim1 = 1
    g1[5] = chunk_len;                            // tensor_dim0_stride
    __builtin_amdgcn_tensor_load_to_lds(g0, g1, g2, g3, g4, 0);
    __builtin_amdgcn_s_wait_tensorcnt((unsigned short)0);
  }
#else
  for (int i = threadIdx.x; i < chunk_len; i += blockDim.x)
    sx[i] = ((const unsigned short*)x)[k0 + i];
#endif
  __syncthreads();

  // ---- Per-group sums of x (shared by all 1024 columns of this block) ----
  for (int g = threadIdx.x; g < ngroups; g += blockDim.x) {
    float s = 0.f;
    #pragma unroll 8
    for (int i = 0; i < GRP; ++i) s += bf1(sx[g * GRP + i]);
    sxsum[g] = s;
  }
  __syncthreads();

  // ---- Stream qweight: 4 columns/thread, coalesced non-temporal b128 ----
  const int n0 = blockIdx.x * COLS_PER_BLOCK + threadIdx.x * COLS;
  const int* qp = qw + (size_t)k0 * NDIM + n0;
  float acc0 = 0.f, acc1 = 0.f, acc2 = 0.f, acc3 = 0.f;

  for (int g = 0; g < ngroups; ++g) {
    const int gabs = (k0 >> 6) + g;
    const uint2 sb = *(const uint2*)((const unsigned short*)sc + (size_t)gabs * NDIM + n0);
    const uint2 zb = *(const uint2*)((const unsigned short*)zr + (size_t)gabs * NDIM + n0);
    const float s0 = bf_lo(sb.x), s1 = bf_hi(sb.x), s2 = bf_lo(sb.y), s3 = bf_hi(sb.y);
    const float z0 = bf_lo(zb.x), z1 = bf_hi(zb.x), z2 = bf_lo(zb.y), z3 = bf_hi(zb.y);

    float a0 = 0.f, a1 = 0.f, a2 = 0.f, a3 = 0.f;
    const unsigned short* xg = &sx[g * GRP];

    #pragma unroll 8
    for (int kk = 0; kk < GRP; kk += 4) {
      const uint2 xb = *(const uint2*)(xg + kk);          // 4 x bf16 values
      const float xk0 = bf_lo(xb.x), xk1 = bf_hi(xb.x);
      const float xk2 = bf_lo(xb.y), xk3 = bf_hi(xb.y);
      // qweight streamed once and never reused: non-temporal (TH_NT) keeps
      // the 192MB L2 for x/scales/zeros/partials.
      const i32x4 qa = __builtin_nontemporal_load((const i32x4*)(qp));
      const i32x4 qb = __builtin_nontemporal_load((const i32x4*)(qp + NDIM));
      const i32x4 qc = __builtin_nontemporal_load((const i32x4*)(qp + 2 * (size_t)NDIM));
      const i32x4 qd = __builtin_nontemporal_load((const i32x4*)(qp + 3 * (size_t)NDIM));
      qp += 4 * (size_t)NDIM;
      a0 = fmaf(xk0, (float)qa[0], a0); a1 = fmaf(xk0, (float)qa[1], a1);
      a2 = fmaf(xk0, (float)qa[2], a2); a3 = fmaf(xk0, (float)qa[3], a3);
      a0 = fmaf(xk1, (float)qb[0], a0); a1 = fmaf(xk1, (float)qb[1], a1);
      a2 = fmaf(xk1, (float)qb[2], a2); a3 = fmaf(xk1, (float)qb[3], a3);
      a0 = fmaf(xk2, (float)qc[0], a0); a1 = fmaf(xk2, (float)qc[1], a1);
      a2 = fmaf(xk2, (float)qc[2], a2); a3 = fmaf(xk2, (float)qc[3], a3);
      a0 = fmaf(xk3, (float)qd[0], a0); a1 = fmaf(xk3, (float)qd[1], a1);
      a2 = fmaf(xk3, (float)qd[2], a2); a3 = fmaf(xk3, (float)qd[3], a3);
    }

    // acc += s * (a - z * sum_x)   (group-factored zero-point)
    const float Xg = sxsum[g];
    acc0 = fmaf(s0, fmaf(-z0, Xg, a0), acc0);
    acc1 = fmaf(s1, fmaf(-z1, Xg, a1), acc1);
    acc2 = fmaf(s2, fmaf(-z2, Xg, a2), acc2);
    acc3 = fmaf(s3, fmaf(-z3, Xg, a3), acc3);
  }

  if (SINGLE) {
    const uint2 bb = *(const uint2*)((const unsigned short*)bias + n0);
    out[n0 + 0] = __float2bfloat16(acc0 + bf_lo(bb.x));
    out[n0 + 1] = __float2bfloat16(acc1 + bf_hi(bb.x));
    out[n0 + 2] = __float2bfloat16(acc2 + bf_lo(bb.y));
    out[n0 + 3] = __float2bfloat16(acc3 + bf_hi(bb.y));
  } else {
    float4 r; r.x = acc0; r.y = acc1; r.z = acc2; r.w = acc3;
    *(float4*)(ws + (size_t)chunk * NDIM + n0) = r;
  }
}

// Split-reduction as a matrix product: out[1,N] = ones[1,16] x ws[16,N] (+bias).
// Each wave reduces 16 columns with 4x V_WMMA_F32_16X16X4_F32 (K=4 each),
// A = 16x4 all-ones, B = 4x16 tile of f32 partials, f32 RNE accumulate.
// B layout (ISA 7.12.2): VGPR0 = rows K={0,2} by lane half, VGPR1 = K={1,3}.
// C row 0 (M=0) = c[0] in lanes 0..15.
__global__ __launch_bounds__(TPB)
void awq_reduce_kernel(const float* __restrict__ ws,
                       const __hip_bfloat16* __restrict__ bias,
                       __hip_bfloat16* __restrict__ out)
{
  const int lane  = threadIdx.x & 31;
  const int wid   = threadIdx.x >> 5;
  const int n     = blockIdx.x * (TPB / 32 * 16) + wid * 16 + (lane & 15);
  const int rbase = (lane >> 4) << 1;   // lanes 0-15 -> rows {0,1}; lanes 16-31 -> {2,3}

  v2f a; a[0] = 1.0f; a[1] = 1.0f;      // 16x4 ones (layout-invariant)
  v8f c = {};
  #pragma unroll
  for (int s = 0; s < SPLIT; s += 4) {
    v2f b;
    b[0] = ws[(size_t)(s + rbase + 0) * NDIM + n];
    b[1] = ws[(size_t)(s + rbase + 1) * NDIM + n];
    c = __builtin_amdgcn_wmma_f32_16x16x4_f32(false, a, false, b,
                                              (short)0, c, false, false);
  }
  if (lane < 16)
    out[n] = __float2bfloat16(c[0] + bf1(((const unsigned short*)bias)[n]));
}

extern "C" void kernel_launch(void* const* d_in, const int* in_sizes, int n_in,
                              void* d_out, int out_size, void* d_ws, size_t ws_size,
                              hipStream_t stream) {
  const __hip_bfloat16* x    = (const __hip_bfloat16*)d_in[0];
  const int*            qw   = (const int*)d_in[1];
  const __hip_bfloat16* sc   = (const __hip_bfloat16*)d_in[2];
  const __hip_bfloat16* zr   = (const __hip_bfloat16*)d_in[3];
  const __hip_bfloat16* bias = (const __hip_bfloat16*)d_in[4];
  __hip_bfloat16*       out  = (__hip_bfloat16*)d_out;

  const size_t need = (size_t)SPLIT * NDIM * sizeof(float);
  if (ws_size >= need) {
    dim3 grid(NB_N, SPLIT);
    awq_gemv_kernel<false><<<grid, TPB, 0, stream>>>(x, qw, sc, zr, bias, out,
                                                     (float*)d_ws, KDIM / SPLIT);
    // 16 cols per wave, 8 waves per block -> 128 cols/block, exact full waves.
    awq_reduce_kernel<<<NDIM / (TPB / 32 * 16), TPB, 0, stream>>>((const float*)d_ws, bias, out);
  } else {
    dim3 grid(NB_N, 1);
    awq_gemv_kernel<true><<<grid, TPB, 0, stream>>>(x, qw, sc, zr, bias, out,
                                                    nullptr, KDIM);
  }
}